// SynEventSampler_21749714387440
// MI455X (gfx1250) — compile-verified
//
#include <hip/hip_runtime.h>
#include <math.h>

#define NUM_SAMPLES 8
#define NUM_EXP 256
#define NUM_SB 40            // NUM_SAMPLES_BOUNDARY
#define NUM_MARK 10
#define BS_TOTAL (32 * 512)  // B*S
#define DTIME_MAX 5.0f
#define OVER_SAMPLE 1.5f

#if __has_builtin(__builtin_amdgcn_tensor_load_to_lds)
#define HAVE_TDM 1
#endif

typedef unsigned int u32x4 __attribute__((ext_vector_type(4)));
typedef int i32x4 __attribute__((ext_vector_type(4)));
typedef int i32x8 __attribute__((ext_vector_type(8)));

// numerically stable softplus: max(x,0) + log1p(exp(-|x|))
__device__ __forceinline__ float softplus_f(float x) {
    float e = __expf(-fabsf(x));
    return fmaxf(x, 0.0f) + __logf(1.0f + e);
}

__global__ __launch_bounds__(256) void syn_event_sampler_kernel(
    const float* __restrict__ time_seq,   // [B,S]
    const float* __restrict__ dt_seq,     // [B,S]
    const int*   __restrict__ event_seq,  // [B,S]
    const float* __restrict__ exp_raw,    // [B,S,E]
    const float* __restrict__ unif,       // [B,S,NS,E]
    const float* __restrict__ emb,        // [20,10]
    const float* __restrict__ scale,      // [10]
    const float* __restrict__ bias,       // [10]
    float*       __restrict__ out)        // res [B,S,NS] ++ weights [B,S,NS]
{
    __shared__ float lds_unif[NUM_SAMPLES * NUM_EXP];  // 8 KB, TDM destination
    __shared__ float lds_bound[NUM_SB];
    __shared__ float lds_part[NUM_SAMPLES * 8];        // per-wave partial mins

    const int bs   = blockIdx.x;     // 0..16383
    const int tid  = threadIdx.x;    // 0..255
    const int lane = tid & 31;
    const int wave = tid >> 5;

#ifdef HAVE_TDM
    // ---- Async tensor DMA: unif tile [8*256 f32, contiguous] -> LDS ----
    // Issued by wave 0 only; it waits for completion before phase 1.
    if (tid == 0) {
        unsigned long long ga =
            (unsigned long long)(uintptr_t)(unif + (size_t)bs * (NUM_SAMPLES * NUM_EXP));
        unsigned int lds_off = (unsigned int)(uintptr_t)(&lds_unif[0]);  // low 32b = LDS byte offset
        const unsigned int nelem = NUM_SAMPLES * NUM_EXP;  // 2048 f32
        u32x4 g0;
        g0[0] = 1u;                                            // count=1, user mode, no gather
        g0[1] = lds_off;                                       // lds_addr (bytes)
        g0[2] = (unsigned int)(ga & 0xffffffffull);            // global_addr[31:0]
        g0[3] = (unsigned int)((ga >> 32) & 0x1ffffffull)      // global_addr[56:32]
              | (2u << 30);                                    // type=2 ("image")
        i32x8 g1;
        g1[0] = (int)(2u << 16);                 // wg_mask=0, data_size=4B
        g1[1] = (int)((nelem & 0xffffu) << 16);  // tensor_dim0[15:0]
        g1[2] = (int)((nelem >> 16) | (1u << 16));   // tensor_dim0[31:16]=0, tensor_dim1=1
        g1[3] = (int)((nelem & 0xffffu) << 16);  // tensor_dim1[31:16]=0, tile_dim0=2048
        g1[4] = 1;                               // tile_dim1=1, tile_dim2=0
        g1[5] = (int)nelem;                      // tensor_dim0_stride[31:0]
        g1[6] = 0;                               // stride0[47:32]=0, stride1[15:0]=0
        g1[7] = 0;
        i32x4 gz = {0, 0, 0, 0};
#if defined(__clang_major__) && __clang_major__ >= 23
        i32x8 gz8 = {0, 0, 0, 0, 0, 0, 0, 0};
        __builtin_amdgcn_tensor_load_to_lds(g0, g1, gz, gz, gz8, 0);
#else
        __builtin_amdgcn_tensor_load_to_lds(g0, g1, gz, gz, 0);
#endif
        __builtin_amdgcn_s_wait_tensorcnt(0);    // s_wait_tensorcnt 0 (wave 0 only)
    }
#endif

    // Uniform per-block scalars (compiler scalarizes: s_load paths)
    const float t   = time_seq[bs];
    const float dtv = dt_seq[bs];
    const int   ev  = event_seq[bs];

    // software prefetch: this thread's candidate draw, needed in phase 3
    const float eraw = exp_raw[(size_t)bs * NUM_EXP + tid];

    float em[NUM_MARK], sc[NUM_MARK];
#pragma unroll
    for (int m = 0; m < NUM_MARK; ++m) {
        em[m] = emb[ev * NUM_MARK + m] + bias[m];
        sc[m] = scale[m];
    }

    // ---- Phase 1: intensity upper bound over 40 boundary points ----
    if (tid < NUM_SB) {
        float tb = t + dtv * ((float)tid * (1.0f / (float)(NUM_SB - 1)));
        float c  = __cosf(tb);
        float s  = 0.0f;
#pragma unroll
        for (int m = 0; m < NUM_MARK; ++m) s += softplus_f(fmaf(sc[m], c, em[m]));
        lds_bound[tid] = s;
    }
    __syncthreads();  // also orders TDM completion (wave0 waited tensorcnt) before LDS reads

    float mx = lds_bound[0];
#pragma unroll 8
    for (int k = 1; k < NUM_SB; ++k) mx = fmaxf(mx, lds_bound[k]);
    const float upper = mx * OVER_SAMPLE;

    // ---- Phase 2/3: candidate time + total intensity for this thread's sample ----
    const float expn = eraw / upper;           // exp_numbers[e]
    const float c2   = __cosf(t + expn);
    float tot = 0.0f;
#pragma unroll
    for (int m = 0; m < NUM_MARK; ++m) tot += softplus_f(fmaf(sc[m], c2, em[m]));
    const float thr = tot / upper;             // accept iff unif < total_int/upper

    // ---- Phase 4: per-slot accept + min-reduce over the 256 candidates ----
    const float INF = __int_as_float(0x7f800000);
#pragma unroll
    for (int ns = 0; ns < NUM_SAMPLES; ++ns) {
#ifdef HAVE_TDM
        float u = lds_unif[ns * NUM_EXP + tid];
#else
        float u = unif[(size_t)bs * (NUM_SAMPLES * NUM_EXP) + ns * NUM_EXP + tid];
#endif
        float v = (u < thr) ? expn : INF;
        // wave32 min-reduction (5 xor-shuffle steps)
#pragma unroll
        for (int off = 16; off >= 1; off >>= 1)
            v = fminf(v, __shfl_xor(v, off, 32));
        if (lane == 0) lds_part[ns * 8 + wave] = v;
    }
    __syncthreads();

    if (tid < NUM_SAMPLES) {
        float v = lds_part[tid * 8];
#pragma unroll
        for (int w = 1; w < 8; ++w) v = fminf(v, lds_part[tid * 8 + w]);
        float res = (v < INF) ? fminf(v, 1e5f) : DTIME_MAX;  // inf -> DTIME_MAX, clamp 1e5
        out[(size_t)bs * NUM_SAMPLES + tid] = res;
        out[(size_t)BS_TOTAL * NUM_SAMPLES + (size_t)bs * NUM_SAMPLES + tid] =
            1.0f / (float)NUM_SAMPLES;  // weights
    }
}

extern "C" void kernel_launch(void* const* d_in, const int* in_sizes, int n_in,
                              void* d_out, int out_size, void* d_ws, size_t ws_size,
                              hipStream_t stream) {
    const float* time_seq = (const float*)d_in[0];
    const float* dt_seq   = (const float*)d_in[1];
    const int*   ev_seq   = (const int*)d_in[2];
    const float* exp_raw  = (const float*)d_in[3];
    const float* unif     = (const float*)d_in[4];
    const float* emb      = (const float*)d_in[5];
    const float* scale    = (const float*)d_in[6];
    const float* bias     = (const float*)d_in[7];
    float* out = (float*)d_out;

    (void)in_sizes; (void)n_in; (void)out_size; (void)d_ws; (void)ws_size;

    syn_event_sampler_kernel<<<dim3(BS_TOTAL), dim3(256), 0, stream>>>(
        time_seq, dt_seq, ev_seq, exp_raw, unif, emb, scale, bias, out);
}